// OutputLayer_15539191677615
// MI455X (gfx1250) — compile-verified
//
#include <hip/hip_runtime.h>

typedef __attribute__((ext_vector_type(2))) float v2f;
typedef __attribute__((ext_vector_type(8))) float v8f;

#define S_LEN 8192
#define D_LEN 4096
#define WAVES 4
#define ROWS 16          // s-rows per block (one 16x16 WMMA tile of output rows)
#define KC   64          // feature-dim chunk staged in LDS per wave iteration
#define LDSS 68          // 64 + 4 pad floats -> conflict-free transposed reads
#define NCH  ((D_LEN / WAVES) / KC)   // 16 chunks per wave

// ---------------------------------------------------------------------------
// Kernel A: dual GEMV  p[mat][s] = sum_d wp[mat][d] * G[mat][s][d]
// via V_WMMA_F32_16X16X4_F32, fed by double-buffered
// GLOBAL_LOAD_ASYNC_TO_LDS_B128 (ASYNCcnt pipeline, no block barriers).
//   A(16x4)  = G tile rows  (M = s-row, K = feature)
//   B(4x16)  = wp chunk broadcast into all 16 N columns
//   D(16x16) = every column holds the same partial p[s0+m]
// ---------------------------------------------------------------------------
__launch_bounds__(WAVES * 32)
__global__ void gemv_wmma_kernel(const float* __restrict__ G1,
                                 const float* __restrict__ G2,
                                 const float* __restrict__ w1,
                                 const float* __restrict__ w2,
                                 float* __restrict__ logits) {
  __shared__ float lds_g[WAVES][2][ROWS * LDSS];  // per-wave double-buffered G tile
  __shared__ float lds_w[WAVES][2][KC];           // per-wave double-buffered wp chunk
  __shared__ float lds_red[WAVES][ROWS];          // cross-wave partial reduce

  const int lane = threadIdx.x & 31;
  const int wave = threadIdx.x >> 5;
  const int blk  = blockIdx.x;
  const int mat  = blk >> 9;              // 0 -> (G_M1, wp1), 1 -> (G_M2, wp2)
  const int s0   = (blk & 511) * ROWS;

  const float* __restrict__ G  = mat ? G2 : G1;
  const float* __restrict__ wp = mat ? w2 : w1;
  const float* __restrict__ Gbase = G + (size_t)s0 * D_LEN;

  const int half = lane >> 4;             // 0: K=0,1  1: K=2,3 (A/B VGPR layout)
  const int rlo  = lane & 15;
  const int dw0  = wave * (D_LEN / WAVES);   // 1024 features per wave

  // --- precompute per-lane async-load coordinates --------------------------
  // Generic->LDS address: low 32 bits of a flat LDS pointer are the LDS byte
  // address (aperture lives in the high 32 bits; ISA 10.2 aperture rules).
  const unsigned lds_g_base = (unsigned)(unsigned long long)&lds_g[wave][0][0];
  const unsigned lds_w_base = (unsigned)(unsigned long long)&lds_w[wave][0][0];

  unsigned voff[8], loff[8];
#pragma unroll
  for (int it = 0; it < 8; ++it) {
    const int idx = it * 32 + lane;       // 0..255
    const int row = idx >> 4;             // 0..15
    const int col = (idx & 15) << 2;      // 0..60, step 4
    voff[it] = (unsigned)((row * D_LEN + col) * 4);        // global byte offset
    loff[it] = lds_g_base + (unsigned)((row * LDSS + col) * 4); // LDS byte addr
  }
  const unsigned wdst  = lds_w_base + (unsigned)(lane * 8);
  const unsigned wvoff = (unsigned)(lane * 8);

  // 9 async loads per chunk: 8 x b128 for the 16x64 G tile, 1 x b64 for wp.
  auto issue_chunk = [&](int dc, int bufsel) {
    const float* sg = Gbase + dc;         // uniform -> SGPR pair (GVS mode)
    const float* sw = wp + dc;
    const unsigned badd = (unsigned)bufsel * (unsigned)(ROWS * LDSS * 4);
    const unsigned wadd = (unsigned)bufsel * (unsigned)(KC * 4);
#pragma unroll
    for (int it = 0; it < 8; ++it) {
      asm volatile("global_load_async_to_lds_b128 %0, %1, %2"
                   :: "v"(loff[it] + badd), "v"(voff[it]), "s"(sg)
                   : "memory");
    }
    asm volatile("global_load_async_to_lds_b64 %0, %1, %2"
                 :: "v"(wdst + wadd), "v"(wvoff), "s"(sw)
                 : "memory");
  };

  v8f c = {0.f, 0.f, 0.f, 0.f, 0.f, 0.f, 0.f, 0.f};

  issue_chunk(dw0, 0);                    // prologue: fill buffer 0

  for (int ci = 0; ci < NCH; ++ci) {
    const int cur = ci & 1;
    if (ci + 1 < NCH) {
      // WAR: make sure our prior ds_reads of the other buffer retired,
      // then prefetch chunk ci+1 into it and wait for chunk ci (<=9 pending).
      asm volatile("s_wait_dscnt 0x0" ::: "memory");
      issue_chunk(dw0 + (ci + 1) * KC, cur ^ 1);
      asm volatile("s_wait_asynccnt 0x9" ::: "memory");
    } else {
      asm volatile("s_wait_asynccnt 0x0" ::: "memory");
    }

    const float* gb = &lds_g[wave][cur][0];
    const float* wb = &lds_w[wave][cur][0];
    // --- 16 WMMAs consume the 64-wide chunk (K = 4 each) -------------------
#pragma unroll
    for (int t = 0; t < 16; ++t) {
      const int kb = 4 * t + (half << 1);
      // A: lane L<16 holds A[L,0..1]; lane 16+L holds A[L,2..3]
      v2f a = *(const v2f*)&gb[rlo * LDSS + kb];
      // B: broadcast wp pair per half-wave (all N columns identical)
      v2f b = *(const v2f*)&wb[kb];
      c = __builtin_amdgcn_wmma_f32_16x16x4_f32(
          /*neg_a=*/false, a, /*neg_b=*/false, b,
          /*c_mod=*/(short)0, c, /*reuse_a=*/false, /*reuse_b=*/false);
    }
  }

  // --- extract column N=0: VGPR j lane0 -> D[j,0], lane16 -> D[j+8,0] ------
  if (lane == 0) {
#pragma unroll
    for (int j = 0; j < 8; ++j) lds_red[wave][j] = c[j];
  }
  if (lane == 16) {
#pragma unroll
    for (int j = 0; j < 8; ++j) lds_red[wave][8 + j] = c[j];
  }
  __syncthreads();

  if (threadIdx.x < ROWS) {
    float sum = 0.f;
#pragma unroll
    for (int w = 0; w < WAVES; ++w) sum += lds_red[w][threadIdx.x];
    logits[mat * S_LEN + s0 + threadIdx.x] = sum;
  }
}

// ---------------------------------------------------------------------------
// Kernel B: row-wise softmax over S=8192 (2 rows). Deterministic fixed-order
// block reductions (no float atomics).
// ---------------------------------------------------------------------------
__launch_bounds__(256)
__global__ void softmax_kernel(const float* __restrict__ logits,
                               float* __restrict__ out) {
  __shared__ float red[256];
  const int row = blockIdx.x;     // 0,1
  const int tid = threadIdx.x;    // 0..255
  const float* __restrict__ x = logits + row * S_LEN;

  float lmax = -3.402823466e+38f;
  for (int i = tid; i < S_LEN; i += 256) lmax = fmaxf(lmax, x[i]);
  red[tid] = lmax;
  __syncthreads();
  for (int s = 128; s > 0; s >>= 1) {
    if (tid < s) red[tid] = fmaxf(red[tid], red[tid + s]);
    __syncthreads();
  }
  const float m = red[0];
  __syncthreads();

  float lsum = 0.f;
  for (int i = tid; i < S_LEN; i += 256) lsum += __expf(x[i] - m);
  red[tid] = lsum;
  __syncthreads();
  for (int s = 128; s > 0; s >>= 1) {
    if (tid < s) red[tid] += red[tid + s];
    __syncthreads();
  }
  const float inv = 1.0f / red[0];

  for (int i = tid; i < S_LEN; i += 256)
    out[row * S_LEN + i] = __expf(x[i] - m) * inv;
}

// ---------------------------------------------------------------------------
extern "C" void kernel_launch(void* const* d_in, const int* in_sizes, int n_in,
                              void* d_out, int out_size, void* d_ws, size_t ws_size,
                              hipStream_t stream) {
  (void)in_sizes; (void)n_in; (void)out_size; (void)ws_size;
  const float* G1 = (const float*)d_in[0];   // [8192, 4096]
  const float* G2 = (const float*)d_in[1];   // [8192, 4096]
  const float* w1 = (const float*)d_in[2];   // [1, 4096]
  const float* w2 = (const float*)d_in[3];   // [1, 4096]
  float* out    = (float*)d_out;             // [1, 2, 8192] flat
  float* logits = (float*)d_ws;              // 2 * 8192 floats scratch

  // 2 matrices x 512 s-tiles (16 rows each) = 1024 blocks, 4 waves per block
  gemv_wmma_kernel<<<1024, WAVES * 32, 0, stream>>>(G1, G2, w1, w2, logits);
  softmax_kernel<<<2, 256, 0, stream>>>(logits, out);
}